// HeteroGNN_6691559047510
// MI455X (gfx1250) — compile-verified
//
#include <hip/hip_runtime.h>

typedef __attribute__((ext_vector_type(2))) float v2f;
typedef __attribute__((ext_vector_type(8))) float v8f;

#define WMMA_F32_OK __has_builtin(__builtin_amdgcn_wmma_f32_16x16x4_f32)

// ---------------------------------------------------------------------------
// WMMA GEMM: C[M,N] = A[M,K] @ W[K,N] (+ bias) (+= existing C if acc)
// One wave per 16-row strip, computing all NT = N/16 column tiles at once:
// the 16x4 f32 A fragment is loaded once per k-step and reused across NT
// independent v_wmma_f32_16x16x4_f32 accumulation chains.
// A fragment: lanes 0-15 hold rows M=0..15 with K=(0,1); lanes 16-31 K=(2,3).
// B fragment: lane&15 = column, same K split. C/D: lane&15 = col, vgpr i =
// row mbase+i, lane>=16 selects rows +8.
// ---------------------------------------------------------------------------
template <int NT>
__global__ void gemm16_wmma(const float* __restrict__ A, const float* __restrict__ W,
                            const float* __restrict__ bias, float* __restrict__ C,
                            int M, int K, int strips, int acc)
{
    constexpr int N = NT * 16;
    int wv = (int)((blockIdx.x * (unsigned)blockDim.x + threadIdx.x) >> 5);
    if (wv >= strips) return;                      // wave-uniform: EXEC stays full
    int lane  = threadIdx.x & 31;
    int l15   = lane & 15;
    int half  = lane >> 4;                         // 0 or 1
    int row   = wv * 16 + l15;                     // A row owned by this lane
    int mbase = wv * 16 + half * 8;                // D rows mbase..mbase+7

    v8f c[NT];
#pragma unroll
    for (int t = 0; t < NT; ++t) c[t] = (v8f){};
    if (acc) {
#pragma unroll
        for (int t = 0; t < NT; ++t) {
            const float* cp = C + (long)mbase * N + t * 16 + l15;
#pragma unroll
            for (int i = 0; i < 8; ++i) c[t][i] = cp[(long)i * N];
        }
    }

    const float* ap = A + (long)row * K + half * 2;    // this lane's A elements
    const float* wb = W + (long)(half * 2) * N + l15;  // this lane's B column base

#if WMMA_F32_OK
    for (int kk = 0; kk < K; kk += 4) {
        float2 av = *(const float2*)(ap + kk);
        v2f a;
        a.x = av.x; a.y = av.y;
        const float* wr = wb + (long)kk * N;           // row kk+half*2 of W
#pragma unroll
        for (int t = 0; t < NT; ++t) {
            v2f b;
            b.x = wr[t * 16];
            b.y = wr[N + t * 16];
            c[t] = __builtin_amdgcn_wmma_f32_16x16x4_f32(false, a, false, b,
                                                         (short)0, c[t], false, false);
        }
    }
#else
    for (int kk = 0; kk < K; ++kk) {
#pragma unroll
        for (int t = 0; t < NT; ++t) {
            float wv2 = W[(long)kk * N + t * 16 + l15];
#pragma unroll
            for (int i = 0; i < 8; ++i) c[t][i] += A[(long)(mbase + i) * K + kk] * wv2;
        }
    }
#endif

#pragma unroll
    for (int t = 0; t < NT; ++t) {
        int col = t * 16 + l15;
        float bv = bias ? bias[col] : 0.0f;
        float* cp = C + (long)mbase * N + col;
#pragma unroll
        for (int i = 0; i < 8; ++i) cp[(long)i * N] = c[t][i] + bv;
    }
}

// ---------------------------------------------------------------------------
// Elementwise / edge kernels
// ---------------------------------------------------------------------------
__global__ void zero_f32(float* p, long n)
{
    long t = blockIdx.x * (long)blockDim.x + threadIdx.x;
    long s = (long)gridDim.x * blockDim.x;
    for (; t < n; t += s) p[t] = 0.0f;
}

__global__ void edge_count(const int* __restrict__ dst, long E, float* cnt)
{
    long t = blockIdx.x * (long)blockDim.x + threadIdx.x;
    if (t >= E) return;
    atomicAdd(&cnt[dst[t]], 1.0f);
}

__global__ void compute_dis(float* deg_io, int n)   // dis = rsqrt(deg + 1)
{
    int i = blockIdx.x * blockDim.x + threadIdx.x;
    if (i >= n) return;
    deg_io[i] = rsqrtf(deg_io[i] + 1.0f);
}

// out[dst,0:64] += X[src,0:64] * (use_norm ? dis[src]*dis[dst] : 1)
__global__ void scatter64(const float* __restrict__ X, const float* __restrict__ dis,
                          const int* __restrict__ src, const int* __restrict__ dst,
                          long E, float* out, int use_norm)
{
    long t = blockIdx.x * (long)blockDim.x + threadIdx.x;
    if (t >= E * 16) return;
    long e = t >> 4;
    int  f = (int)(t & 15) * 4;
    int  s = src[e], d = dst[e];
    float nrm = use_norm ? dis[s] * dis[d] : 1.0f;
    float4 v = *(const float4*)(X + (long)s * 64 + f);
    float* op = out + (long)d * 64 + f;
    atomicAdd(op + 0, v.x * nrm);
    atomicAdd(op + 1, v.y * nrm);
    atomicAdd(op + 2, v.z * nrm);
    atomicAdd(op + 3, v.w * nrm);
}

__global__ void mean_div(float* sum, const float* __restrict__ cnt, long n64)
{
    long t = blockIdx.x * (long)blockDim.x + threadIdx.x;
    if (t >= n64) return;
    sum[t] = sum[t] / fmaxf(cnt[t >> 6], 1.0f);
}

__global__ void rowdot64(const float* __restrict__ X, const float* __restrict__ a,
                         float* out, int n)
{
    int i = blockIdx.x * blockDim.x + threadIdx.x;
    if (i >= n) return;
    const float4* xr = (const float4*)(X + (long)i * 64);
    const float4* ar = (const float4*)a;
    float s = 0.0f;
#pragma unroll
    for (int j = 0; j < 16; ++j) {
        float4 v = xr[j], w = ar[j];
        s += v.x * w.x + v.y * w.y + v.z * w.z + v.w * w.w;
    }
    out[i] = s;
}

__device__ inline unsigned enc_f(float f)
{
    int i = __float_as_int(f);
    return (i >= 0) ? ((unsigned)i | 0x80000000u) : ~(unsigned)i;
}
__device__ inline float dec_f(unsigned u)
{
    return (u & 0x80000000u) ? __int_as_float((int)(u & 0x7fffffffu))
                             : __int_as_float((int)~u);
}

__global__ void gat_edge_e(const float* __restrict__ as_, const float* __restrict__ ad_,
                           const int* __restrict__ src, const int* __restrict__ dst,
                           long E, float* ebuf, unsigned* menc)
{
    long t = blockIdx.x * (long)blockDim.x + threadIdx.x;
    if (t >= E) return;
    float e = as_[src[t]] + ad_[dst[t]];
    e = e > 0.0f ? e : 0.2f * e;                   // leaky ReLU, slope 0.2
    ebuf[t] = e;
    atomicMax(&menc[dst[t]], enc_f(e));
}

__global__ void decode_max(unsigned* m, int n)     // in-place -> float
{
    int i = blockIdx.x * blockDim.x + threadIdx.x;
    if (i >= n) return;
    unsigned u = m[i];
    float f = (u == 0u) ? 0.0f : dec_f(u);         // u==0 <=> no in-edges
    ((float*)m)[i] = f;
}

__global__ void gat_ex(float* e_io, const float* __restrict__ m,
                       const int* __restrict__ dst, long E, float* denom)
{
    long t = blockIdx.x * (long)blockDim.x + threadIdx.x;
    if (t >= E) return;
    float ex = expf(e_io[t] - m[dst[t]]);
    e_io[t] = ex;
    atomicAdd(&denom[dst[t]], ex);
}

__global__ void gat_scatter(const float* __restrict__ ex, const float* __restrict__ denom,
                            const float* __restrict__ xs,
                            const int* __restrict__ src, const int* __restrict__ dst,
                            long E, float* out)
{
    long t = blockIdx.x * (long)blockDim.x + threadIdx.x;
    if (t >= E * 16) return;
    long e = t >> 4;
    int  f = (int)(t & 15) * 4;
    int  s = src[e], d = dst[e];
    float w = ex[e] / denom[d];
    float4 v = *(const float4*)(xs + (long)s * 64 + f);
    float* op = out + (long)d * 64 + f;
    atomicAdd(op + 0, v.x * w);
    atomicAdd(op + 1, v.y * w);
    atomicAdd(op + 2, v.z * w);
    atomicAdd(op + 3, v.w * w);
}

// xp_new = relu(agg + xl*dis^2 + gb[f])
__global__ void combine_paper(const float* __restrict__ agg, const float* __restrict__ xl,
                              const float* __restrict__ dis, const float* __restrict__ gb,
                              float* out, long n64)
{
    long t = blockIdx.x * (long)blockDim.x + threadIdx.x;
    if (t >= n64) return;
    float di = dis[t >> 6];
    float v = agg[t] + xl[t] * di * di + gb[t & 63];
    out[t] = v > 0.0f ? v : 0.0f;
}

__global__ void bias_relu64(float* io, const float* __restrict__ b, long n64)
{
    long t = blockIdx.x * (long)blockDim.x + threadIdx.x;
    if (t >= n64) return;
    float v = io[t] + b[t & 63];
    io[t] = v > 0.0f ? v : 0.0f;
}

// ---------------------------------------------------------------------------
// Launch
// ---------------------------------------------------------------------------
extern "C" void kernel_launch(void* const* d_in, const int* in_sizes, int n_in,
                              void* d_out, int out_size, void* d_ws, size_t ws_size,
                              hipStream_t stream)
{
    const int DP = 128, DA = 64, H = 64, OUTC = 32;
    const float* x_paper  = (const float*)d_in[0];
    const float* x_author = (const float*)d_in[1];
    const float* gW0  = (const float*)d_in[2];
    const float* gb0  = (const float*)d_in[3];
    const float* sWl0 = (const float*)d_in[4];
    const float* sbl0 = (const float*)d_in[5];
    const float* sWr0 = (const float*)d_in[6];
    const float* aWs0 = (const float*)d_in[7];
    const float* aWd0 = (const float*)d_in[8];
    const float* aas0 = (const float*)d_in[9];
    const float* aad0 = (const float*)d_in[10];
    const float* ab0  = (const float*)d_in[11];
    // layer-2 GCN/SAGE params (indices 12..16) are dead code: x_paper after
    // layer 2 never reaches the output.
    const float* aWs1 = (const float*)d_in[17];
    const float* aWd1 = (const float*)d_in[18];
    const float* aas1 = (const float*)d_in[19];
    const float* aad1 = (const float*)d_in[20];
    const float* ab1  = (const float*)d_in[21];
    const float* linW = (const float*)d_in[22];
    const float* linb = (const float*)d_in[23];
    const int* csrc = (const int*)d_in[24];
    const int* cdst = (const int*)d_in[25];
    const int* wsrc = (const int*)d_in[26];
    const int* wdst = (const int*)d_in[27];

    const long EC = in_sizes[24], EW = in_sizes[26];
    const int  NP = in_sizes[0] / DP, NA = in_sizes[1] / DA;
    const long NP64 = (long)NP * 64, NA64 = (long)NA * 64;

    // ---- carve workspace ----
    char* w = (char*)d_ws;
    auto carve = [&](size_t bytes) { void* p = (void*)w; w += (bytes + 255) & ~(size_t)255; return p; };
    float*    xp1  = (float*)carve(NP64 * 4);      // x_paper after layer 1
    float*    xl   = (float*)carve(NP64 * 4);      // GCN xl / layer-2 xs
    float*    agg  = (float*)carve(NP64 * 4);      // GCN edge agg + SAGE accum
    float*    sum  = (float*)carve(NP64 * 4);      // SAGE sum/mean, then layer-1 xs
    float*    xd   = (float*)carve(NA64 * 4);      // GAT dst transform
    float*    xa1  = (float*)carve(NA64 * 4);      // x_author after layer 1
    float*    xa2  = (float*)carve(NA64 * 4);      // x_author after layer 2
    float*    dis  = (float*)carve((size_t)NP * 4);
    float*    cnt  = (float*)carve((size_t)NP * 4);
    float*    asrc = (float*)carve((size_t)NP * 4);
    float*    adst = (float*)carve((size_t)NA * 4);
    unsigned* menc = (unsigned*)carve((size_t)NA * 4);
    float*    den  = (float*)carve((size_t)NA * 4);
    float*    ebuf = (float*)carve((size_t)EW * 4);

    auto grd = [](long n) { return (int)((n + 255) / 256); };
    auto Z   = [&](float* p, long n) { zero_f32<<<2048, 256, 0, stream>>>(p, n); };
    auto gemm = [&](const float* A, const float* W_, const float* bias, float* C,
                    int M, int K, int N, int acc) {
        int strips = M / 16;                       // one wave per 16-row strip
        int blocks = grd((long)strips * 32);
        if (N == 64)
            gemm16_wmma<4><<<blocks, 256, 0, stream>>>(A, W_, bias, C, M, K, strips, acc);
        else
            gemm16_wmma<2><<<blocks, 256, 0, stream>>>(A, W_, bias, C, M, K, strips, acc);
    };

    // ================= Layer 1 =================
    Z(dis, NP); Z(cnt, NP); Z(agg, NP64); Z(sum, NP64);
    Z((float*)menc, NA); Z(den, NA); Z(xa1, NA64);

    // GCN (cites: paper->paper)
    edge_count<<<grd(EC), 256, 0, stream>>>(cdst, EC, dis);
    compute_dis<<<grd(NP), 256, 0, stream>>>(dis, NP);
    gemm(x_paper, gW0, nullptr, xl, NP, DP, H, 0);
    scatter64<<<grd(EC * 16), 256, 0, stream>>>(xl, dis, csrc, cdst, EC, agg, 1);

    // SAGE (writes: author->paper)
    edge_count<<<grd(EW), 256, 0, stream>>>(wdst, EW, cnt);
    scatter64<<<grd(EW * 16), 256, 0, stream>>>(x_author, nullptr, wsrc, wdst, EW, sum, 0);
    mean_div<<<grd(NP64), 256, 0, stream>>>(sum, cnt, NP64);
    gemm(sum, sWl0, sbl0, agg, NP, H, H, 1);        // agg += mean @ Wl + bl
    gemm(x_paper, sWr0, nullptr, agg, NP, DP, H, 1);// agg += x_paper @ Wr
    combine_paper<<<grd(NP64), 256, 0, stream>>>(agg, xl, dis, gb0, xp1, NP64);

    // GAT (rev writes: paper->author); src=wdst (papers), dst=wsrc (authors)
    gemm(x_paper, aWs0, nullptr, sum, NP, DP, H, 0);   // xs (reuse sum)
    gemm(x_author, aWd0, nullptr, xd, NA, DA, H, 0);   // xd
    rowdot64<<<grd(NP), 256, 0, stream>>>(sum, aas0, asrc, NP);
    rowdot64<<<grd(NA), 256, 0, stream>>>(xd, aad0, adst, NA);
    gat_edge_e<<<grd(EW), 256, 0, stream>>>(asrc, adst, wdst, wsrc, EW, ebuf, menc);
    decode_max<<<grd(NA), 256, 0, stream>>>(menc, NA);
    gat_ex<<<grd(EW), 256, 0, stream>>>(ebuf, (const float*)menc, wsrc, EW, den);
    gat_scatter<<<grd(EW * 16), 256, 0, stream>>>(ebuf, den, sum, wdst, wsrc, EW, xa1);
    bias_relu64<<<grd(NA64), 256, 0, stream>>>(xa1, ab0, NA64);

    // ================= Layer 2 (GAT only; paper path is dead) =================
    Z((float*)menc, NA); Z(den, NA); Z(xa2, NA64);
    gemm(xp1, aWs1, nullptr, xl, NP, H, H, 0);         // xs2 (reuse xl)
    gemm(xa1, aWd1, nullptr, xd, NA, H, H, 0);
    rowdot64<<<grd(NP), 256, 0, stream>>>(xl, aas1, asrc, NP);
    rowdot64<<<grd(NA), 256, 0, stream>>>(xd, aad1, adst, NA);
    gat_edge_e<<<grd(EW), 256, 0, stream>>>(asrc, adst, wdst, wsrc, EW, ebuf, menc);
    decode_max<<<grd(NA), 256, 0, stream>>>(menc, NA);
    gat_ex<<<grd(EW), 256, 0, stream>>>(ebuf, (const float*)menc, wsrc, EW, den);
    gat_scatter<<<grd(EW * 16), 256, 0, stream>>>(ebuf, den, xl, wdst, wsrc, EW, xa2);
    bias_relu64<<<grd(NA64), 256, 0, stream>>>(xa2, ab1, NA64);

    // ================= Final linear =================
    gemm(xa2, linW, linb, (float*)d_out, NA, H, OUTC, 0);

    (void)n_in; (void)out_size; (void)ws_size;
}